// Hash_2010044695129
// MI455X (gfx1250) — compile-verified
//
#include <hip/hip_runtime.h>
#include <stdint.h>

// ---------------------------------------------------------------------------
// Instant-NGP multiresolution hash encoding, MI455X (gfx1250, wave32).
//
// Roofline: 67M random 8B gathers from 64MB of tables (L2-resident on the
// 192MB L2), 64MB streamed output, ~0.9G VALU ops -> gather-bound, not
// compute-bound. WMMA is inapplicable (per-row B operand in the weighted
// reduction), so the CDNA5 wins are: TDM async tile load of coordinates into
// LDS (tensor_load_to_lds + s_wait_tensorcnt), wave32-coalesced output
// layout (lane = level), and non-temporal stores to keep L2 for the tables.
// ---------------------------------------------------------------------------

#define NLEVELS    16
#define LOG2_T     19
#define TMASK      ((1u << LOG2_T) - 1u)
#define PRIME_Y    2654435761u
#define PRIME_Z    805459861u

#define TILE_PTS   64                       // points per block tile
#define BLOCK      256                      // 8 waves
#define PAIRS      (TILE_PTS * NLEVELS)     // 1024 (point,level) pairs / block
#define ITERS      (PAIRS / BLOCK)          // 4

typedef float    f32x2 __attribute__((ext_vector_type(2)));
typedef unsigned u32x4 __attribute__((ext_vector_type(4)));
typedef int      i32x8 __attribute__((ext_vector_type(8)));
typedef int      i32x4 __attribute__((ext_vector_type(4)));

__global__ __launch_bounds__(BLOCK) void hashgrid_fwd_kernel(
    const float* __restrict__ x,        // [N,3]
    const float* __restrict__ tables,   // [16, 2^19, 2]
    float*       __restrict__ out,      // [N, 32]
    int n_points)
{
    __shared__ float sx[TILE_PTS * 3];

    const int base_pt = blockIdx.x * TILE_PTS;
    int pts = n_points - base_pt;
    if (pts > TILE_PTS) pts = TILE_PTS;
    const int K = pts * 3;                      // floats in this x tile

    // ---- Stage the coordinate tile into LDS via the Tensor Data Mover ----
#if defined(__gfx1250__) && __has_builtin(__builtin_amdgcn_tensor_load_to_lds)
    if (threadIdx.x < 32) {                     // wave 0 issues the DMA
        const unsigned long long ga =
            (unsigned long long)(uintptr_t)(x + (size_t)base_pt * 3);
        // Flat shared-aperture address: bits[31:0] are the LDS byte offset.
        const unsigned lds_off = (unsigned)(uintptr_t)(void*)sx;

        // D# group0: count=1 | lds_addr | global_addr[56:0] | type=2
        u32x4 g0 = { 1u,
                     lds_off,
                     (unsigned)(ga & 0xFFFFFFFFu),
                     (unsigned)((ga >> 32) & 0x1FFFFFFu) | (2u << 30) };
        // D# group1: wg_mask=0 (not in cluster), data_size=4B,
        //   tensor_dim0=K, tensor_dim1=1, tile_dim0=K, tile_dim1=1,
        //   tile_dim2=0, dim0_stride=K, dim1_stride=K
        i32x8 g1 = { (int)(2u << 16),                       // data_size=2 (4B)
                     (int)((unsigned)(K & 0xFFFF) << 16),   // tensor_dim0[15:0]
                     (int)((unsigned)(K >> 16) | (1u << 16)),// dim0[31:16]|dim1 lo
                     (int)((unsigned)(K & 0xFFFF) << 16),   // dim1 hi=0 | tile_dim0
                     1,                                     // tile_dim1=1, tile_dim2=0
                     K,                                     // dim0_stride[31:0]
                     (int)((unsigned)(K & 0xFFFF) << 16),   // stride hi=0 | dim1_stride lo
                     0 };
        i32x4 gz4 = { 0, 0, 0, 0 };                         // <=2D: groups 2/3 unused
        i32x8 gz8 = { 0, 0, 0, 0, 0, 0, 0, 0 };            // extra group (clang-23 6-arg form)
        __builtin_amdgcn_tensor_load_to_lds(g0, g1, gz4, gz4, gz8, 0);
        __builtin_amdgcn_s_wait_tensorcnt(0);
    }
    __syncthreads();
#else
    for (int i = threadIdx.x; i < K; i += BLOCK)
        sx[i] = x[(size_t)base_pt * 3 + i];
    __syncthreads();
#endif

    // ---- Per-thread invariants: level = tid & 15 (wave32-coalesced out) ---
    const int   l   = threadIdx.x & (NLEVELS - 1);
    // res_l = floor(16 * b^l), b = 128^(1/15) = 2^(7/15)
    const float res = floorf(16.0f * exp2f((float)l * (7.0f / 15.0f)));
    const f32x2* __restrict__ tab =
        (const f32x2*)(tables + ((size_t)l << (LOG2_T + 1)));

#pragma unroll
    for (int it = 0; it < ITERS; ++it) {
        const int pair = it * BLOCK + (int)threadIdx.x;   // 0..1023
        const int lp   = pair >> 4;                       // local point
        const int n    = base_pt + lp;
        if (n >= n_points) continue;

        const float px = sx[lp * 3 + 0] * res;
        const float py = sx[lp * 3 + 1] * res;
        const float pz = sx[lp * 3 + 2] * res;
        const float x0 = floorf(px), y0 = floorf(py), z0 = floorf(pz);
        const float rx = px - x0,    ry = py - y0,    rz = pz - z0;

        const unsigned ix = (unsigned)x0, iy = (unsigned)y0, iz = (unsigned)z0;
        // h(p+1) = h(p) + prime  (mod 2^32) -> 2 mults total for 8 corners
        const unsigned hx0 = ix,              hx1 = ix + 1u;
        const unsigned hy0 = iy * PRIME_Y,    hy1 = hy0 + PRIME_Y;
        const unsigned hz0 = iz * PRIME_Z,    hz1 = hz0 + PRIME_Z;

        const unsigned i000 = (hx0 ^ hy0 ^ hz0) & TMASK;
        const unsigned i100 = (hx1 ^ hy0 ^ hz0) & TMASK;
        const unsigned i010 = (hx0 ^ hy1 ^ hz0) & TMASK;
        const unsigned i110 = (hx1 ^ hy1 ^ hz0) & TMASK;
        const unsigned i001 = (hx0 ^ hy0 ^ hz1) & TMASK;
        const unsigned i101 = (hx1 ^ hy0 ^ hz1) & TMASK;
        const unsigned i011 = (hx0 ^ hy1 ^ hz1) & TMASK;
        const unsigned i111 = (hx1 ^ hy1 ^ hz1) & TMASK;

        // 8 independent global_load_b64 gathers (compiler clauses them)
        const f32x2 f000 = tab[i000];
        const f32x2 f100 = tab[i100];
        const f32x2 f010 = tab[i010];
        const f32x2 f110 = tab[i110];
        const f32x2 f001 = tab[i001];
        const f32x2 f101 = tab[i101];
        const f32x2 f011 = tab[i011];
        const f32x2 f111 = tab[i111];

        const float wx0 = 1.0f - rx, wy0 = 1.0f - ry, wz0 = 1.0f - rz;
        const float w000 = wx0 * wy0 * wz0, w100 = rx * wy0 * wz0;
        const float w010 = wx0 * ry  * wz0, w110 = rx * ry  * wz0;
        const float w001 = wx0 * wy0 * rz,  w101 = rx * wy0 * rz;
        const float w011 = wx0 * ry  * rz,  w111 = rx * ry  * rz;

        f32x2 acc;
        acc.x = w000 * f000.x;                 acc.y = w000 * f000.y;
        acc.x = fmaf(w100, f100.x, acc.x);     acc.y = fmaf(w100, f100.y, acc.y);
        acc.x = fmaf(w010, f010.x, acc.x);     acc.y = fmaf(w010, f010.y, acc.y);
        acc.x = fmaf(w110, f110.x, acc.x);     acc.y = fmaf(w110, f110.y, acc.y);
        acc.x = fmaf(w001, f001.x, acc.x);     acc.y = fmaf(w001, f001.y, acc.y);
        acc.x = fmaf(w101, f101.x, acc.x);     acc.y = fmaf(w101, f101.y, acc.y);
        acc.x = fmaf(w011, f011.x, acc.x);     acc.y = fmaf(w011, f011.y, acc.y);
        acc.x = fmaf(w111, f111.x, acc.x);     acc.y = fmaf(w111, f111.y, acc.y);

        // out[n, 2l .. 2l+1]; lanes of a wave cover one contiguous 256B span.
        // Non-temporal: stream past L2 so table lines stay resident.
        f32x2* op = (f32x2*)(out + (size_t)n * (NLEVELS * 2) + (size_t)l * 2);
        __builtin_nontemporal_store(acc, op);
    }
}

extern "C" void kernel_launch(void* const* d_in, const int* in_sizes, int n_in,
                              void* d_out, int out_size, void* d_ws, size_t ws_size,
                              hipStream_t stream)
{
    const float* x      = (const float*)d_in[0];   // [N,3] float32
    const float* tables = (const float*)d_in[1];   // [16, 2^19, 2] float32
    float*       out    = (float*)d_out;           // [N, 32] float32

    const int n_points = in_sizes[0] / 3;
    const int grid     = (n_points + TILE_PTS - 1) / TILE_PTS;
    hashgrid_fwd_kernel<<<grid, BLOCK, 0, stream>>>(x, tables, out, n_points);
}